// GATModel_90460601188377
// MI455X (gfx1250) — compile-verified
//
#include <hip/hip_runtime.h>
#include <hip/hip_bf16.h>

typedef __attribute__((ext_vector_type(2))) float v2f;
typedef __attribute__((ext_vector_type(8))) float v8f;
typedef int v4i_vs __attribute__((vector_size(16)));   // matches builtin's expected pointee

#define IN_FEATS 128
#define HEADS 4
#define D_HEAD 32
#define NEG_SLOPE 0.2f

#if defined(__has_builtin)
#if __has_builtin(__builtin_amdgcn_global_load_async_to_lds_b128)
#define HAVE_ASYNC_LDS 1
#endif
#endif

// ---------- helpers: order-preserving float<->uint for atomicMax ----------
__device__ __forceinline__ unsigned fenc(float f) {
    unsigned u = __float_as_uint(f);
    return (u & 0x80000000u) ? ~u : (u | 0x80000000u);
}
__device__ __forceinline__ float fdec(unsigned k) {
    if (k == 0u) return 0.0f;  // sentinel: no edge touched this node/head
    unsigned u = (k & 0x80000000u) ? (k & 0x7FFFFFFFu) : ~k;
    return __uint_as_float(u);
}

// ---------- 0: zero d_out (rst accumulator), denom, emax keys ----------
__global__ void gat_init(float* __restrict__ out, float* __restrict__ denom,
                         unsigned* __restrict__ emaxk, int n_out, int n_nh) {
    int i = blockIdx.x * blockDim.x + threadIdx.x;
    if (i < n_out) out[i] = 0.0f;
    if (i < n_nh) { denom[i] = 0.0f; emaxk[i] = 0u; }
}

// ---------- 1: feat = features @ W via V_WMMA_F32_16X16X4_F32 ----------
// grid.x = N/16 row strips; 8 waves per block cover the 8 column tiles of 128.
// A strip staged into LDS via async DMA (ASYNCcnt) when available.
__global__ void __launch_bounds__(256)
gat_gemm_wmma(const float* __restrict__ A, const float* __restrict__ B,
              float* __restrict__ C, int N) {
    __shared__ float As[16 * 132];   // 16 rows x 128, padded stride 132 (LDS bank spread)
    const int row0 = blockIdx.x * 16;
    const int lane = threadIdx.x & 31;
    const int wave = threadIdx.x >> 5;      // 0..7 -> column tile

#ifdef HAVE_ASYNC_LDS
    // async global -> LDS DMA, 16B per lane, no VGPR round-trip
    {
        for (int i = threadIdx.x; i < 16 * 32; i += 256) {
            int r  = i >> 5;     // row 0..15
            int c4 = i & 31;     // float4 index 0..31
            __builtin_amdgcn_global_load_async_to_lds_b128(
                (v4i_vs*)(const_cast<float*>(A) + (size_t)(row0 + r) * IN_FEATS + c4 * 4),
                (v4i_vs*)&As[r * 132 + c4 * 4],
                /*offset=*/0, /*cpol=*/0);
        }
#if __has_builtin(__builtin_amdgcn_s_wait_asynccnt)
        __builtin_amdgcn_s_wait_asynccnt(0);
#else
        asm volatile("s_wait_asynccnt 0x0" ::: "memory");
#endif
    }
    __syncthreads();
#else
    // fallback: cooperative coalesced load through VGPRs
    for (int i = threadIdx.x; i < 16 * 32; i += 256) {
        int r  = i >> 5;
        int c4 = i & 31;
        float4 v = ((const float4*)(A + (size_t)(row0 + r) * IN_FEATS))[c4];
        *(float4*)&As[r * 132 + c4 * 4] = v;
    }
    __syncthreads();
#endif

    const int m   = lane & 15;              // row within tile (A) / col within tile (B,D)
    const int kb  = (lane >> 4) << 1;       // K sub-offset: lanes 0-15 -> 0, 16-31 -> 2
    const int col = wave * 16 + m;

    v8f acc = {};
    #pragma unroll
    for (int k0 = 0; k0 < IN_FEATS; k0 += 4) {
        v2f a, b;
        a.x = As[m * 132 + k0 + kb];
        a.y = As[m * 132 + k0 + kb + 1];
        b.x = B[(size_t)(k0 + kb) * IN_FEATS + col];
        b.y = B[(size_t)(k0 + kb + 1) * IN_FEATS + col];
        acc = __builtin_amdgcn_wmma_f32_16x16x4_f32(
            /*neg_a=*/false, a, /*neg_b=*/false, b,
            /*c_mod=*/(short)0, acc, /*reuse_a=*/false, /*reuse_b=*/false);
    }

    const int mrow = (lane >> 4) * 8;       // lanes 16-31 hold M=8..15
    #pragma unroll
    for (int j = 0; j < 8; ++j)
        C[(size_t)(row0 + mrow + j) * IN_FEATS + col] = acc[j];
}

// ---------- 2: per-node attention logits el/er ----------
__global__ void gat_node_logits(const float* __restrict__ feat,
                                const float* __restrict__ attn_l,
                                const float* __restrict__ attn_r,
                                float* __restrict__ el, float* __restrict__ er,
                                int n_nh) {
    int i = blockIdx.x * blockDim.x + threadIdx.x;
    if (i >= n_nh) return;
    int n = i >> 2, h = i & 3;
    const float* f  = feat + (size_t)n * IN_FEATS + h * D_HEAD;
    const float* al = attn_l + h * D_HEAD;
    const float* ar = attn_r + h * D_HEAD;
    float sl = 0.0f, sr = 0.0f;
    #pragma unroll
    for (int d = 0; d < D_HEAD; ++d) {
        float v = f[d];
        sl += v * al[d];
        sr += v * ar[d];
    }
    el[i] = sl;
    er[i] = sr;
}

// ---------- 3: edge logits + leaky_relu + segment max (atomicMax on keys) ----------
__global__ void gat_edge_logits(const int* __restrict__ src, const int* __restrict__ dst,
                                const float* __restrict__ el, const float* __restrict__ er,
                                float* __restrict__ eW, unsigned* __restrict__ emaxk, int E) {
    int i = blockIdx.x * blockDim.x + threadIdx.x;
    if (i >= E) return;
    int s = src[i], d = dst[i];
    float4 a = *(const float4*)(el + (size_t)s * HEADS);
    float4 b = *(const float4*)(er + (size_t)d * HEADS);
    float e0 = a.x + b.x, e1 = a.y + b.y, e2 = a.z + b.z, e3 = a.w + b.w;
    e0 = e0 > 0.0f ? e0 : NEG_SLOPE * e0;
    e1 = e1 > 0.0f ? e1 : NEG_SLOPE * e1;
    e2 = e2 > 0.0f ? e2 : NEG_SLOPE * e2;
    e3 = e3 > 0.0f ? e3 : NEG_SLOPE * e3;
    *(float4*)(eW + (size_t)i * HEADS) = make_float4(e0, e1, e2, e3);
    unsigned* k = emaxk + (size_t)d * HEADS;
    atomicMax(&k[0], fenc(e0));
    atomicMax(&k[1], fenc(e1));
    atomicMax(&k[2], fenc(e2));
    atomicMax(&k[3], fenc(e3));
}

// ---------- 4: ee = exp(e - emax[dst]); denom[dst] += ee ----------
__global__ void gat_edge_exp(const int* __restrict__ dst,
                             float* __restrict__ eW,           // in: e, out: ee
                             const unsigned* __restrict__ emaxk,
                             float* __restrict__ denom, int E) {
    int i = blockIdx.x * blockDim.x + threadIdx.x;
    if (i >= E) return;
    int d = dst[i];
    float4 e4 = *(const float4*)(eW + (size_t)i * HEADS);
    const unsigned* k = emaxk + (size_t)d * HEADS;
    float ee0 = __expf(e4.x - fdec(k[0]));
    float ee1 = __expf(e4.y - fdec(k[1]));
    float ee2 = __expf(e4.z - fdec(k[2]));
    float ee3 = __expf(e4.w - fdec(k[3]));
    *(float4*)(eW + (size_t)i * HEADS) = make_float4(ee0, ee1, ee2, ee3);
    float* dn = denom + (size_t)d * HEADS;
    atomicAdd(&dn[0], ee0);
    atomicAdd(&dn[1], ee1);
    atomicAdd(&dn[2], ee2);
    atomicAdd(&dn[3], ee3);
}

// ---------- 5: wave-per-edge weighted scatter-add aggregation ----------
// lane l handles feature h*32 + l for h = 0..3; fully coalesced 128B lane groups.
__global__ void __launch_bounds__(256)
gat_aggregate(const int* __restrict__ src, const int* __restrict__ dst,
              const float* __restrict__ feat, const float* __restrict__ eW,
              const float* __restrict__ denom, float* __restrict__ out, int E) {
    int gid  = blockIdx.x * blockDim.x + threadIdx.x;
    int edge = gid >> 5;
    int lane = gid & 31;
    if (edge >= E) return;
    int s = src[edge], d = dst[edge];
    const float* fs = feat + (size_t)s * IN_FEATS;
    float*       od = out  + (size_t)d * IN_FEATS;
    const float* ee = eW   + (size_t)edge * HEADS;
    const float* dn = denom + (size_t)d * HEADS;
    #pragma unroll
    for (int h = 0; h < HEADS; ++h) {
        float alpha = ee[h] / fmaxf(dn[h], 1e-9f);   // same for all lanes -> broadcast
        atomicAdd(&od[h * D_HEAD + lane], alpha * fs[h * D_HEAD + lane]);
    }
}

// ---------- 6: epilogue: out = elu(out + bias) ----------
__global__ void gat_elu(float* __restrict__ out, const float* __restrict__ bias, int n) {
    int i = blockIdx.x * blockDim.x + threadIdx.x;
    if (i >= n) return;
    float x = out[i] + bias[i & (IN_FEATS - 1)];
    out[i] = x > 0.0f ? x : (__expf(x) - 1.0f);
}

extern "C" void kernel_launch(void* const* d_in, const int* in_sizes, int n_in,
                              void* d_out, int out_size, void* d_ws, size_t ws_size,
                              hipStream_t stream) {
    const float* features = (const float*)d_in[0];
    const int*   src      = (const int*)  d_in[1];
    const int*   dst      = (const int*)  d_in[2];
    const float* W        = (const float*)d_in[3];
    const float* attn_l   = (const float*)d_in[4];
    const float* attn_r   = (const float*)d_in[5];
    const float* bias     = (const float*)d_in[6];

    const int N = in_sizes[0] / IN_FEATS;   // 100000
    const int E = in_sizes[1];              // 1600000
    const int n_out = N * IN_FEATS;         // 12.8M floats
    const int n_nh  = N * HEADS;            // 400k

    float* out = (float*)d_out;

    // workspace layout (all 16B aligned)
    float*    feat  = (float*)d_ws;
    float*    el    = feat  + (size_t)n_out;
    float*    er    = el    + (size_t)n_nh;
    float*    denom = er    + (size_t)n_nh;
    float*    eW    = denom + (size_t)n_nh;
    unsigned* emaxk = (unsigned*)(eW + (size_t)E * HEADS);

    const int TB = 256;

    // 0: init accumulators
    gat_init<<<(n_out + TB - 1) / TB, TB, 0, stream>>>(out, denom, emaxk, n_out, n_nh);

    // 1: projection GEMM (WMMA f32)
    gat_gemm_wmma<<<N / 16, TB, 0, stream>>>(features, W, feat, N);

    // 2: node attention logits
    gat_node_logits<<<(n_nh + TB - 1) / TB, TB, 0, stream>>>(feat, attn_l, attn_r, el, er, n_nh);

    // 3: edge logits + segment max
    gat_edge_logits<<<(E + TB - 1) / TB, TB, 0, stream>>>(src, dst, el, er, eW, emaxk, E);

    // 4: exp + segment sum
    gat_edge_exp<<<(E + TB - 1) / TB, TB, 0, stream>>>(dst, eW, emaxk, denom, E);

    // 5: weighted scatter-add aggregation (1 wave per edge)
    long long agg_threads = (long long)E * 32;
    gat_aggregate<<<(int)((agg_threads + TB - 1) / TB), TB, 0, stream>>>(src, dst, feat, eW, denom, out, E);

    // 6: bias + ELU
    gat_elu<<<(n_out + TB - 1) / TB, TB, 0, stream>>>(out, bias, n_out);
}